// ImportanceAggregator_28424093564971
// MI455X (gfx1250) — compile-verified
//
#include <hip/hip_runtime.h>
#include <hip/hip_bf16.h>

// CDNA5 / gfx1250 fused importance-aggregation kernel.
//   x[n]   = sum_k wnorm[n,k] * features[nbr[n,k]]      (gather, LDS tile)
//   agg[n] = x[n] @ W + b                               (V_WMMA_F32_16X16X4_F32)
//   out[n] = LayerNorm(agg[n]) * gamma + beta           (fused, shfl reductions)
//
// 48 rows/block (3 WMMA M-tiles) amortizes the per-block W stream (256KB from
// L2) 3x vs a single M-tile, and gives each B fragment 3 WMMA uses per load.

typedef __attribute__((ext_vector_type(2))) float v2f;
typedef __attribute__((ext_vector_type(8))) float v8f;

#define DMODEL 256
#define KNBR   16
#define MT     3               // M tiles per block
#define ROWS   (MT * 16)       // 48 nodes per workgroup
#define LDSP   260             // padded LDS row stride (floats): 260%64==4 -> conflict-free

__global__ __launch_bounds__(256)
void importance_agg_ln_kernel(const float* __restrict__ feat,      // [N,256]
                              const long long* __restrict__ nbrs,  // [N,16] int64
                              const float* __restrict__ iw,        // [N,16]
                              const float* __restrict__ W,         // [256,256] row-major (x @ W)
                              const float* __restrict__ bias,      // [256]
                              const float* __restrict__ gamma,     // [256]
                              const float* __restrict__ beta,      // [256]
                              float* __restrict__ out,             // [N,256]
                              int N)
{
    __shared__ float lds[ROWS * LDSP];   // phase1: x tile; phase3: agg tile (~50.3KB)
    __shared__ float s_mean[ROWS];
    __shared__ float s_rstd[ROWS];

    const int t    = threadIdx.x;        // 0..255
    const int base = blockIdx.x * ROWS;

    // ---------------- Phase 1: weighted gather-aggregate into LDS ----------------
    // thread t owns feature dim t; loop nodes serially, 16 independent loads per k-loop.
    for (int ni = 0; ni < ROWS; ++ni) {
        int node = base + ni;
        if (node >= N) node = N - 1;     // duplicate last row for tail (stores guarded later)
        const float*     wrow = iw   + (long long)node * KNBR;
        const long long* nrow = nbrs + (long long)node * KNBR;

        float wsum = 0.0f;
        #pragma unroll
        for (int k = 0; k < KNBR; ++k) wsum += wrow[k];
        const bool  zsum = (wsum == 0.0f);
        const float invw = zsum ? 0.0f : (1.0f / wsum);

        float acc = 0.0f;
        #pragma unroll
        for (int k = 0; k < KNBR; ++k) {
            const long long nb = nrow[k];
            if (k + 1 < KNBR) {
                // hint next gather row toward cache (global_prefetch_b8)
                __builtin_prefetch(feat + (long long)nrow[k + 1] * DMODEL + t, 0, 1);
            }
            const float wn = zsum ? (1.0f / KNBR) : (wrow[k] * invw);
            acc = fmaf(wn, feat[(long long)nb * DMODEL + t], acc);
        }
        lds[ni * LDSP + t] = acc;
    }
    __syncthreads();

    // ---------------- Phase 2: fp32 WMMA GEMM: agg[48x256] = x @ W ----------------
    const int wave  = t >> 5;            // 0..7
    const int lane  = t & 31;
    const int lhalf = lane >> 4;         // 0: lanes 0-15, 1: lanes 16-31
    const int l16   = lane & 15;

    const int col0 = (wave * 2 + 0) * 16 + l16;     // two 16-wide column tiles per wave
    const int col1 = (wave * 2 + 1) * 16 + l16;

    v8f acc[MT][2];
    #pragma unroll
    for (int m = 0; m < MT; ++m) { acc[m][0] = (v8f){}; acc[m][1] = (v8f){}; }

    // A 16x4 f32 layout: lanes 0-15 hold M=lane, K=k0/k0+1; lanes 16-31 hold K=k0+2/k0+3.
    // B 4x16 f32 layout mirrors: VGPR0/1 = rows k0(+2)/k0+1(+3), N striped on lanes.
    #pragma unroll 4
    for (int kb = 0; kb < DMODEL / 4; ++kb) {
        const int k0 = kb * 4 + lhalf * 2;

        v2f b0, b1;
        b0.x = W[(k0 + 0) * DMODEL + col0];
        b0.y = W[(k0 + 1) * DMODEL + col0];
        b1.x = W[(k0 + 0) * DMODEL + col1];
        b1.y = W[(k0 + 1) * DMODEL + col1];

        #pragma unroll
        for (int m = 0; m < MT; ++m) {
            v2f a;
            a.x = lds[(m * 16 + l16) * LDSP + k0];
            a.y = lds[(m * 16 + l16) * LDSP + k0 + 1];
            acc[m][0] = __builtin_amdgcn_wmma_f32_16x16x4_f32(false, a, false, b0,
                                                              (short)0, acc[m][0], false, false);
            acc[m][1] = __builtin_amdgcn_wmma_f32_16x16x4_f32(false, a, false, b1,
                                                              (short)0, acc[m][1], false, false);
        }
    }
    __syncthreads();   // x tile fully consumed; reuse LDS for agg tile

    // C/D layout: VGPR r -> row r + 8*lhalf (within M tile), col = tile*16 + l16
    const float bv0 = bias[col0];
    const float bv1 = bias[col1];
    #pragma unroll
    for (int m = 0; m < MT; ++m) {
        #pragma unroll
        for (int r = 0; r < 8; ++r) {
            const int row = m * 16 + r + lhalf * 8;
            lds[row * LDSP + col0] = acc[m][0][r] + bv0;
            lds[row * LDSP + col1] = acc[m][1][r] + bv1;
        }
    }
    __syncthreads();

    // ---------------- Phase 3: fused LayerNorm over the 256 features ----------------
    {
        const int c16 = t & 15;          // 16 lanes cooperate per row
        #pragma unroll
        for (int rg = 0; rg < MT; ++rg) {
            const int r = rg * 16 + (t >> 4);
            float s = 0.0f, s2 = 0.0f;
            #pragma unroll
            for (int j = 0; j < 16; ++j) {
                const float v = lds[r * LDSP + j * 16 + c16];
                s  += v;
                s2 += v * v;
            }
            #pragma unroll
            for (int off = 8; off >= 1; off >>= 1) {
                s  += __shfl_xor(s,  off, 16);
                s2 += __shfl_xor(s2, off, 16);
            }
            if (c16 == 0) {
                const float mean = s * (1.0f / DMODEL);
                const float var  = s2 * (1.0f / DMODEL) - mean * mean;
                s_mean[r] = mean;
                s_rstd[r] = rsqrtf(var + 1e-5f);
            }
        }
    }
    __syncthreads();

    // coalesced normalized store: 48 contiguous 1KB row-chunks
    #pragma unroll 4
    for (int j = 0; j < ROWS; ++j) {
        const int idx  = j * 256 + t;    // over the 48x256 tile
        const int row  = idx >> 8;
        const int col  = idx & 255;
        const int node = base + row;
        if (node < N) {
            const float v = lds[row * LDSP + col];
            out[(long long)node * DMODEL + col] =
                (v - s_mean[row]) * s_rstd[row] * gamma[col] + beta[col];
        }
    }
}

extern "C" void kernel_launch(void* const* d_in, const int* in_sizes, int n_in,
                              void* d_out, int out_size, void* d_ws, size_t ws_size,
                              hipStream_t stream) {
    const float*     feat  = (const float*)d_in[0];      // [N,256] f32
    const long long* nbrs  = (const long long*)d_in[1];  // [N,16]  i64
    const float*     iw    = (const float*)d_in[2];      // [N,16]  f32
    const float*     W     = (const float*)d_in[3];      // [256,256] f32
    const float*     bias  = (const float*)d_in[4];      // [256] f32
    const float*     gamma = (const float*)d_in[5];      // [256] f32
    const float*     beta  = (const float*)d_in[6];      // [256] f32
    float*           out   = (float*)d_out;

    const int N = in_sizes[0] / DMODEL;                  // 50000
    const int nblocks = (N + ROWS - 1) / ROWS;           // 1042

    importance_agg_ln_kernel<<<nblocks, 256, 0, stream>>>(
        feat, nbrs, iw, W, bias, gamma, beta, out, N);
}